// BalanceLoss_67379446940206
// MI455X (gfx1250) — compile-verified
//
#include <hip/hip_runtime.h>

typedef __attribute__((ext_vector_type(2))) float v2f;
typedef __attribute__((ext_vector_type(4))) float v4f;
typedef __attribute__((ext_vector_type(8))) float v8f;

#define NBINS 2048
#define LMAX 13.8155106f           // -log(1e-6), max possible BCE term
#define LN2  0.69314718055994531f
#define FXS  4294967296.0          // 2^32 fixed-point scale

// Workspace layout (as u64*):
//   [0      .. 2048)  : u64 bin_sum_fx[NBINS]
//   [2048   .. 3072)  : u32 bin_cnt[NBINS]   (1024 u64 slots)
//   [3072+0]          : u64 pos_cnt
//   [3072+1]          : u64 neg_cnt
//   [3072+2]          : u64 pos_loss_fx  (sum of gt*mask*loss, *2^32)
//   [3072+3]          : u64 tot_loss_fx  (sum of mask*loss,    *2^32)
#define WS_U32_WORDS ((3072 * 8 + 32) / 4)

__global__ void bl_zero(unsigned* __restrict__ ws, int n) {
    int i = blockIdx.x * blockDim.x + threadIdx.x;
    if (i < n) ws[i] = 0u;
}

__global__ __launch_bounds__(256) void bl_main(const float* __restrict__ pred,
                                               const float* __restrict__ gt,
                                               const float* __restrict__ mask,
                                               unsigned long long* __restrict__ ws,
                                               int N) {
    unsigned long long* gbin_sum = ws;
    unsigned*           gbin_cnt = (unsigned*)(ws + NBINS);
    unsigned long long* gscal    = ws + NBINS + NBINS / 2;

    __shared__ unsigned           lcnt[NBINS];   //  8 KB
    __shared__ unsigned long long lsum[NBINS];   // 16 KB
    for (int b = threadIdx.x; b < NBINS; b += blockDim.x) { lcnt[b] = 0u; lsum[b] = 0ull; }
    __syncthreads();

    const int tid    = blockIdx.x * blockDim.x + threadIdx.x;
    const int stride = gridDim.x * blockDim.x;
    const int n4     = N >> 2;
    const int nfull  = (n4 / stride) * stride;   // uniform trip count region (EXEC all-ones)

    const v4f* p4 = (const v4f*)pred;
    const v4f* g4 = (const v4f*)gt;
    const v4f* m4 = (const v4f*)mask;

    v8f cp = {};                 // WMMA accumulator: sum of gt*mask*loss
    v8f ct = {};                 // WMMA accumulator: sum of mask*loss
    const v2f ones = {1.0f, 1.0f};
    float posc = 0.0f, mc = 0.0f;     // integral-valued float counters (exact)
    float ep = 0.0f, et = 0.0f;       // non-WMMA spillover accumulators

    auto elem = [&](float p, float g, float m, float& pv, float& tv) {
        // gt is exactly 0/1 and pred is clipped to [1e-6, 1-1e-6]: raw v_log_f32
        // (log2) + one multiply is sufficient — no denorm/extended-precision path.
        float sel = (g > 0.5f) ? p : 1.0f - p;
        float l   = __builtin_amdgcn_logf(sel) * (-LN2);   // = -log(sel)
        float pm  = g * m;
        posc += pm;
        mc   += m;
        pv = pm * l;
        tv = m * l;
        if (m > 0.5f && g < 0.5f) {                        // hard-negative histogram
            int b = (int)(l * ((float)NBINS / LMAX));
            b = b < 0 ? 0 : (b > NBINS - 1 ? NBINS - 1 : b);
            atomicAdd(&lcnt[b], 1u);
            atomicAdd(&lsum[b], (unsigned long long)(l * (float)FXS));
        }
    };

    // Main streaming loop: 48B/iter/lane in (non-temporal: zero-reuse stream,
    // keep the 192MB L2 clean), 64 f32 summed per WMMA via ones-matmul.
    for (int i = tid; i < nfull; i += stride) {
        v4f p = __builtin_nontemporal_load(&p4[i]);
        v4f g = __builtin_nontemporal_load(&g4[i]);
        v4f m = __builtin_nontemporal_load(&m4[i]);
        float pv0, tv0, pv1, tv1, pv2, tv2, pv3, tv3;
        elem(p.x, g.x, m.x, pv0, tv0);
        elem(p.y, g.y, m.y, pv1, tv1);
        elem(p.z, g.z, m.z, pv2, tv2);
        elem(p.w, g.w, m.w, pv3, tv3);
        v2f ap = {pv0 + pv1, pv2 + pv3};
        v2f at = {tv0 + tv1, tv2 + tv3};
        // D[M][N] += sum_K A[M][K]  (B = ones) : matrix-pipe reduction, C chains.
        cp = __builtin_amdgcn_wmma_f32_16x16x4_f32(false, ap, false, ones, (short)0, cp, false, false);
        ct = __builtin_amdgcn_wmma_f32_16x16x4_f32(false, at, false, ones, (short)0, ct, false, false);
    }
    // Non-uniform remainder (divergent -> no WMMA here).
    for (int i = nfull + tid; i < n4; i += stride) {
        v4f p = p4[i], g = g4[i], m = m4[i];
        float pv, tv;
        elem(p.x, g.x, m.x, pv, tv); ep += pv; et += tv;
        elem(p.y, g.y, m.y, pv, tv); ep += pv; et += tv;
        elem(p.z, g.z, m.z, pv, tv); ep += pv; et += tv;
        elem(p.w, g.w, m.w, pv, tv); ep += pv; et += tv;
    }
    for (int i = (n4 << 2) + tid; i < N; i += stride) {   // scalar tail
        float pv, tv;
        elem(pred[i], gt[i], mask[i], pv, tv); ep += pv; et += tv;
    }

    // Collapse WMMA C: 8 VGPRs -> half-row sums; xor16 -> full wave sum (all lanes).
    float sp = cp[0] + cp[1] + cp[2] + cp[3] + cp[4] + cp[5] + cp[6] + cp[7];
    float st = ct[0] + ct[1] + ct[2] + ct[3] + ct[4] + ct[5] + ct[6] + ct[7];
    sp += __shfl_xor(sp, 16, 32);
    st += __shfl_xor(st, 16, 32);
    for (int o = 16; o > 0; o >>= 1) {
        ep   += __shfl_xor(ep, o, 32);
        et   += __shfl_xor(et, o, 32);
        posc += __shfl_xor(posc, o, 32);
        mc   += __shfl_xor(mc, o, 32);
    }
    sp += ep;
    st += et;
    if ((threadIdx.x & 31) == 0) {   // deterministic integer/fixed-point atomics
        atomicAdd(&gscal[0], (unsigned long long)(posc + 0.5f));
        atomicAdd(&gscal[1], (unsigned long long)((mc - posc) + 0.5f));
        atomicAdd(&gscal[2], (unsigned long long)((double)sp * FXS));
        atomicAdd(&gscal[3], (unsigned long long)((double)st * FXS));
    }
    __syncthreads();
    for (int b = threadIdx.x; b < NBINS; b += blockDim.x) {
        unsigned c = lcnt[b];
        if (c) {
            atomicAdd(&gbin_cnt[b], c);
            atomicAdd(&gbin_sum[b], lsum[b]);
        }
    }
}

__global__ __launch_bounds__(1024) void bl_select(const unsigned long long* __restrict__ ws,
                                                  float* __restrict__ out) {
    const unsigned long long* gbin_sum = ws;
    const unsigned*           gbin_cnt = (const unsigned*)(ws + NBINS);
    const unsigned long long* gscal    = ws + NBINS + NBINS / 2;

    __shared__ unsigned sc[NBINS];           // suffix-count scan buffer (reversed order)
    __shared__ int bstar;
    __shared__ unsigned long long Shi;
    const int t = threadIdx.x;
    if (t == 0) { bstar = -1; Shi = 0ull; }
    sc[t]        = gbin_cnt[NBINS - 1 - t];
    sc[t + 1024] = gbin_cnt[NBINS - 1 - (t + 1024)];
    __syncthreads();
    for (int off = 1; off < NBINS; off <<= 1) {          // inclusive Hillis-Steele scan
        unsigned v0 = (t >= off) ? sc[t - off] : 0u;
        unsigned v1 = ((t + 1024) >= off) ? sc[t + 1024 - off] : 0u;
        __syncthreads();
        sc[t] += v0;
        sc[t + 1024] += v1;
        __syncthreads();
    }
    // C(b) = #neg with bin >= b  ==  sc[NBINS-1-b]

    unsigned long long posc   = gscal[0], negc = gscal[1];
    unsigned long long pos_fx = gscal[2], tot_fx = gscal[3];
    double pos_loss     = (double)pos_fx * (1.0 / FXS);
    double tot_loss     = (double)tot_fx * (1.0 / FXS);
    double neg_loss_all = tot_loss - pos_loss;
    unsigned long long K = (negc < 3ull * posc) ? negc : 3ull * posc;

    double topk = 0.0;
    if (K > 0 && K < negc) {                 // uniform branch: histogram selection
        for (int b = t; b < NBINS; b += 1024) {
            unsigned Cb  = sc[NBINS - 1 - b];
            unsigned Cb1 = (b == NBINS - 1) ? 0u : sc[NBINS - 2 - b];
            if ((unsigned long long)Cb1 < K && K <= (unsigned long long)Cb) bstar = b;
        }
        __syncthreads();
        int bs = bstar;
        unsigned long long local = 0ull;
        for (int b = t; b < NBINS; b += 1024)
            if (b > bs) local += gbin_sum[b];
        if (local) atomicAdd(&Shi, local);
        __syncthreads();
        unsigned Cb1            = (bs >= NBINS - 1) ? 0u : sc[NBINS - 2 - bs];
        unsigned long long rem  = K - (unsigned long long)Cb1;
        unsigned cb             = (bs >= 0) ? gbin_cnt[bs] : 0u;
        double avg = cb ? ((double)gbin_sum[bs] * (1.0 / FXS)) / (double)cb : 0.0;
        topk = (double)Shi * (1.0 / FXS) + (double)rem * avg;
    } else if (K >= negc && K > 0) {
        topk = neg_loss_all;                 // top-K == all negatives: exact
    }

    if (t == 0) {
        double result;
        if (K > 0) result = (pos_loss + topk) / ((double)posc + (double)K + 1e-6);
        else       result = pos_loss / ((double)posc + 1e-6);
        out[0] = (float)result;
    }
}

extern "C" void kernel_launch(void* const* d_in, const int* in_sizes, int n_in,
                              void* d_out, int out_size, void* d_ws, size_t ws_size,
                              hipStream_t stream) {
    (void)n_in; (void)out_size; (void)ws_size;
    const float* pred = (const float*)d_in[0];
    const float* gt   = (const float*)d_in[1];
    const float* mask = (const float*)d_in[2];
    const int N = in_sizes[0];
    unsigned long long* ws = (unsigned long long*)d_ws;

    bl_zero<<<(WS_U32_WORDS + 255) / 256, 256, 0, stream>>>((unsigned*)d_ws, WS_U32_WORDS);
    bl_main<<<512, 256, 0, stream>>>(pred, gt, mask, ws, N);
    bl_select<<<1, 1024, 0, stream>>>(ws, (float*)d_out);
}